// ErosionP4_76390288326904
// MI455X (gfx1250) — compile-verified
//
#include <hip/hip_runtime.h>
#include <hip/hip_bf16.h>
#include <stdint.h>

// Problem constants (from reference)
#define BB 4
#define GG 4
#define HH 96
#define WW 96
#define CC 4
#define KHH 5
#define KWW 5
#define FF 8

// Tiling
#define TH 8             // output rows per workgroup
#define LROWS (TH + 4)   // 12 padded rows staged in LDS
#define LCOLS (WW + 4)   // 100 = 96 interior + 2+2 inf pad columns
#define NTHREADS 256

typedef unsigned int v4u __attribute__((ext_vector_type(4)));
typedef int          v8i __attribute__((ext_vector_type(8)));
typedef int          v4i __attribute__((ext_vector_type(4)));

// krev[g,dy,dx,k] -> flattened (i,j,depth) index into kernel[5][5][3][.][.]
// Derived from: k_ero = {rot270(K[:,:,2]), K[:,:,1], rot90(K[:,:,0])};
// krot[g]=rot90^g(k_ero); krev = krot[:, ::-1, ::-1].
__device__ __forceinline__ int widx_fn(int g, int k, int dy, int dx) {
    int p, q;
    if      (g == 0) { p = 4 - dy; q = 4 - dx; }
    else if (g == 1) { p = 4 - dx; q = dy;     }
    else if (g == 2) { p = dy;     q = dx;     }
    else             { p = dx;     q = 4 - dy; }
    int i, j, d;
    if      (k == 0) { i = 4 - q; j = p;     d = 2; }
    else if (k == 1) { i = p;     j = q;     d = 1; }
    else             { i = q;     j = 4 - p; d = 0; }
    return (i * 5 + j) * 3 + d;   // *(CC*FF) gives float offset
}

__global__ __launch_bounds__(NTHREADS, 1)
void erosion_p4_kernel(const float* __restrict__ x,
                       const float* __restrict__ kern,
                       float* __restrict__ out) {
    // LDS: xs[k_plane][c][row][col], per-c deinterleaved, +inf padded borders
    __shared__ float xs[3 * CC * LROWS * LCOLS];

    const int tid = threadIdx.x;
    const int h0  = blockIdx.x * TH;
    const int g   = blockIdx.y;
    const int b   = blockIdx.z;
    const float INF = __builtin_inff();

    // ---- 1) fill entire staging buffer with +inf (erosion SAME padding) ----
    for (int i = tid; i < 3 * CC * LROWS * LCOLS; i += NTHREADS) xs[i] = INF;
    __syncthreads();

    // valid input row range for this tile
    const int rlo   = (h0 - 2 < 0) ? 0 : (h0 - 2);
    const int rhi   = (h0 + TH + 2 > HH) ? HH : (h0 + TH + 2);
    const int nrows = rhi - rlo;
    const int r0off = rlo - (h0 - 2);   // 2 at top edge, else 0

#if defined(__has_builtin) && __has_builtin(__builtin_amdgcn_tensor_load_to_lds)
    // ---- 2) TDM: 12 descriptors (3 k-planes x 4 channels), wave 0 only ----
    // Each: tile_dim0 = 1 elem, tile_dim1 = 96 lines with stride 4 elems
    // (deinterleaves [w][c] -> c-plane row), iterated nrows times with
    // lds_addr_increment = 100 elems (padded LDS row stride) and
    // global_addr_increment = 384 elems (one x row).
    if (tid < 32) {
        for (int k = 0; k < 3; ++k) {
            const int gi = (g + k + 3) & 3;          // (g + k - 1) mod 4
            for (int c = 0; c < CC; ++c) {
                const uint64_t ga = (uint64_t)(uintptr_t)
                    (x + ((((size_t)(b * GG + gi) * HH + rlo) * WW) * CC + c));
                const uint32_t la = (uint32_t)(uintptr_t)xs +
                    (uint32_t)((((k * CC + c) * LROWS + r0off) * LCOLS + 2) * 4);
                // D# group 0: count=1 | lds_addr | global_addr | type=2
                v4u g0 = { 1u, la, (uint32_t)ga,
                           ((uint32_t)(ga >> 32) & 0x01FFFFFFu) | (2u << 30) };
                // D# group 1: data_size=4B(2), iterate_enable=1,
                // tensor_dim0 = tensor_dim1 = 1<<20 (no clipping),
                // tile_dim0=1, tile_dim1=96, tensor_dim0_stride=4
                v8i g1 = { (int)((2u << 16) | (1u << 19)),
                           0, 0x10, 0x00010010, WW, CC, 0, 0 };
                // D# group 2 (iterate): tensor_dim2=1, lds_inc=100,
                // global_inc=384, iterate_count = nrows-1
                v4i g2 = { 1, LCOLS, WW * CC, (nrows - 1) << 16 };
                v4i g3 = { 0, 0, 0, 0 };
                v8i g4 = { 0, 0, 0, 0, 0, 0, 0, 0 };
                // clang-23 / therock lane: 6-arg form
                __builtin_amdgcn_tensor_load_to_lds(g0, g1, g2, g3, g4, 0);
            }
        }
#if __has_builtin(__builtin_amdgcn_s_wait_tensorcnt)
        __builtin_amdgcn_s_wait_tensorcnt(0);
#else
        asm volatile("s_wait_tensorcnt 0" ::: "memory");
#endif
    }
#else
    // ---- fallback: cooperative vector-load staging ----
    {
        const int total = 3 * CC * nrows * WW;
        for (int i = tid; i < total; i += NTHREADS) {
            int w = i % WW; int t = i / WW;
            int rr = t % nrows; t /= nrows;
            int c = t % CC; int k = t / CC;
            int gi = (g + k + 3) & 3;
            xs[((k * CC + c) * LROWS + r0off + rr) * LCOLS + 2 + w] =
                x[(((size_t)(b * GG + gi) * HH + (rlo + rr)) * WW + w) * CC + c];
        }
    }
#endif
    __syncthreads();

    // ---- 3) compute: one row per wave, 3 column chunks of 32 ----
    const int lane = tid & 31;
    const int ly   = tid >> 5;           // 0..7 (wave32 -> 8 waves)

#pragma unroll 1
    for (int pi = 0; pi < 3; ++pi) {
        const int lx = lane + pi * 32;   // 0..95
        float sum[FF];
#pragma unroll
        for (int f = 0; f < FF; ++f) sum[f] = 0.0f;

#pragma unroll 1
        for (int c = 0; c < CC; ++c) {
            float acc[FF];
#pragma unroll
            for (int f = 0; f < FF; ++f) acc[f] = INF;

#pragma unroll 1
            for (int k = 0; k < 3; ++k) {
#pragma unroll 1
                for (int dy = 0; dy < KHH; ++dy) {
                    const float* xrow =
                        &xs[((k * CC + c) * LROWS + ly + dy) * LCOLS + lx];
#pragma unroll
                    for (int dx = 0; dx < KWW; ++dx) {
                        const float v = xrow[dx];                 // ds_load_b32
                        const int widx = widx_fn(g, k, dy, dx);   // uniform SALU
                        const float* wp = kern + widx * (CC * FF) + c * FF;
#pragma unroll
                        for (int f = 0; f < FF; ++f)
                            acc[f] = fminf(acc[f], v - wp[f]);    // wp uniform
                    }
                }
            }
#pragma unroll
            for (int f = 0; f < FF; ++f) sum[f] += acc[f];
        }

        const size_t ob =
            ((((size_t)b * GG + g) * HH + (h0 + ly)) * WW + lx) * FF;
        float4* op = (float4*)(out + ob);
        op[0] = make_float4(sum[0], sum[1], sum[2], sum[3]);
        op[1] = make_float4(sum[4], sum[5], sum[6], sum[7]);
    }
}

extern "C" void kernel_launch(void* const* d_in, const int* in_sizes, int n_in,
                              void* d_out, int out_size, void* d_ws, size_t ws_size,
                              hipStream_t stream) {
    (void)in_sizes; (void)n_in; (void)out_size; (void)d_ws; (void)ws_size;
    const float* x    = (const float*)d_in[0];
    const float* kern = (const float*)d_in[1];
    float* out        = (float*)d_out;

    dim3 grid(HH / TH, GG, BB);   // 12 x 4 x 4 = 192 workgroups
    dim3 block(NTHREADS);         // 8 wave32s
    hipLaunchKernelGGL(erosion_p4_kernel, grid, block, 0, stream, x, kern, out);
}